// TransformerNet_13932873908889
// MI455X (gfx1250) — compile-verified
//
#include <hip/hip_runtime.h>
#include <math.h>

// ---------------------------------------------------------------------------
// Problem constants (from reference)
// ---------------------------------------------------------------------------
#define NN 8192      // nodes
#define EE 32768     // edges
#define HH 8         // heads
#define GG 64        // graphs
#define OUTD 10

typedef float v2f __attribute__((ext_vector_type(2)));
typedef float v8f __attribute__((ext_vector_type(8)));

// Order-preserving float <-> uint transform for atomicMax-based segment max.
__device__ __forceinline__ unsigned f2ord(float f) {
  unsigned u = __float_as_uint(f);
  return (u & 0x80000000u) ? ~u : (u | 0x80000000u);
}
__device__ __forceinline__ float ord2f(unsigned u) {
  unsigned b = (u & 0x80000000u) ? (u ^ 0x80000000u) : ~u;
  return __uint_as_float(b);
}

__global__ void __launch_bounds__(256) zero_f32(float* p, int n) {
  int i = blockIdx.x * 256 + threadIdx.x;
  if (i < n) p[i] = 0.0f;
}

// ---------------------------------------------------------------------------
// fp32 GEMM via V_WMMA_F32_16X16X4_F32 (wave32).
// D[M,Nc] = A[M,K] @ B[K,Nc] + bias (bias may be nullptr).
// One wave owns a 16(M) x 64(N) macro tile (4 WMMA accumulators, A reused).
// Block = 256 threads = 8 waves stacked along M -> block tile 128 x 64.
// Requires M%128==0, Nc%64==0, K%4==0 (true for all shapes here).
//
// fp32 A 16x4 lane layout (ISA 7.12.2): lanes 0-15 hold K=k0,k0+1 of row M=lane;
// lanes 16-31 hold K=k0+2,k0+3.  B 4x16: VGPR0 = rows k0 (lanes 0-15) / k0+1
// (lanes 16-31), VGPR1 = rows k0+2 / k0+3, N = lane%16.
// ---------------------------------------------------------------------------
__global__ void __launch_bounds__(256) gemm_wmma_f32(
    const float* __restrict__ A, const float* __restrict__ B,
    const float* __restrict__ bias, float* __restrict__ D,
    int M, int Nc, int K) {
  const int lane  = threadIdx.x & 31;
  const int wid   = threadIdx.x >> 5;
  const int hi    = lane >> 4;       // 0: lanes 0-15, 1: lanes 16-31
  const int nl    = lane & 15;
  const int tileM = blockIdx.y * 8 + wid;
  const int n0    = blockIdx.x * 64;
  const int mrow  = tileM * 16 + nl;

  v8f acc0 = {0,0,0,0,0,0,0,0};
  v8f acc1 = acc0, acc2 = acc0, acc3 = acc0;

  const float* ap = A + (size_t)mrow * K + hi * 2;
  for (int k0 = 0; k0 < K; k0 += 4) {
    v2f a; a.x = ap[k0]; a.y = ap[k0 + 1];
    const float* bp  = B + (size_t)(k0 + hi) * Nc + n0 + nl;
    const float* bp2 = bp + (size_t)2 * Nc;
    v2f b0, b1, b2, b3;
    b0.x = bp[0];  b0.y = bp2[0];
    b1.x = bp[16]; b1.y = bp2[16];
    b2.x = bp[32]; b2.y = bp2[32];
    b3.x = bp[48]; b3.y = bp2[48];
    acc0 = __builtin_amdgcn_wmma_f32_16x16x4_f32(false, a, false, b0, (short)0, acc0, false, false);
    acc1 = __builtin_amdgcn_wmma_f32_16x16x4_f32(false, a, false, b1, (short)0, acc1, false, false);
    acc2 = __builtin_amdgcn_wmma_f32_16x16x4_f32(false, a, false, b2, (short)0, acc2, false, false);
    acc3 = __builtin_amdgcn_wmma_f32_16x16x4_f32(false, a, false, b3, (short)0, acc3, false, false);
  }

  float bb0 = bias ? bias[n0 + nl +  0] : 0.f;
  float bb1 = bias ? bias[n0 + nl + 16] : 0.f;
  float bb2 = bias ? bias[n0 + nl + 32] : 0.f;
  float bb3 = bias ? bias[n0 + nl + 48] : 0.f;
  // C/D layout: VGPR r -> M = r (lanes 0-15) or r+8 (lanes 16-31), N = lane%16.
  #pragma unroll
  for (int r = 0; r < 8; ++r) {
    int row = tileM * 16 + r + hi * 8;
    float* dp = D + (size_t)row * Nc + n0 + nl;
    dp[0]  = acc0[r] + bb0;
    dp[16] = acc1[r] + bb1;
    dp[32] = acc2[r] + bb2;
    dp[48] = acc3[r] + bb3;
  }
}

// ---------------------------------------------------------------------------
// Edge attention phase.  Edge embedding e[h,c] = sum_j ea[e,j]*We[j, h*C+c]
// is computed on the fly (We is tiny, L2-resident) -> saves a 512MB buffer.
// One wave per (edge, head); lanes stride the C channels.
// ---------------------------------------------------------------------------
__global__ void __launch_bounds__(256) edge_logits(
    const float* __restrict__ q, const float* __restrict__ k,
    const float* __restrict__ ea, const float* __restrict__ We,
    const int* __restrict__ src, const int* __restrict__ dst,
    float* __restrict__ logits, unsigned* __restrict__ nodemax,
    int C, int HC, float scale) {
  const int lane = threadIdx.x & 31;
  const int gw   = blockIdx.x * 8 + (threadIdx.x >> 5);
  const int e = gw >> 3, h = gw & 7;
  const int s = src[e], d = dst[e];
  const float ea0 = ea[e*4+0], ea1 = ea[e*4+1], ea2 = ea[e*4+2], ea3 = ea[e*4+3];
  const float* qp = q + (size_t)d * HC + h * C;
  const float* kp = k + (size_t)s * HC + h * C;
  const float* wp = We + h * C;
  float part = 0.f;
  for (int c = lane; c < C; c += 32) {
    float ev = ea0*wp[c] + ea1*wp[HC+c] + ea2*wp[2*HC+c] + ea3*wp[3*HC+c];
    part += qp[c] * (kp[c] + ev);
  }
  for (int off = 16; off; off >>= 1) part += __shfl_xor(part, off, 32);
  if (lane == 0) {
    float lg = part * scale;
    logits[e * HH + h] = lg;
    atomicMax(&nodemax[(size_t)d * HH + h], f2ord(lg));
  }
}

__global__ void __launch_bounds__(256) edge_softmax(
    float* __restrict__ logits, const unsigned* __restrict__ nodemax,
    float* __restrict__ denom, const int* __restrict__ dst, int total) {
  int i = blockIdx.x * 256 + threadIdx.x;
  if (i >= total) return;
  int e = i >> 3, h = i & 7;
  int d = dst[e];
  float ex = expf(logits[i] - ord2f(nodemax[d * HH + h]));
  logits[i] = ex;                               // reuse buffer: now holds exp
  atomicAdd(&denom[d * HH + h], ex);
}

__global__ void __launch_bounds__(256) edge_scatter(
    const float* __restrict__ v, const float* __restrict__ ea,
    const float* __restrict__ We, const int* __restrict__ src,
    const int* __restrict__ dst, const float* __restrict__ exbuf,
    const float* __restrict__ denom, float* __restrict__ agg,
    int C, int HC) {
  const int lane = threadIdx.x & 31;
  const int gw   = blockIdx.x * 8 + (threadIdx.x >> 5);
  const int e = gw >> 3, h = gw & 7;
  const int s = src[e], d = dst[e];
  const float alpha = exbuf[e * HH + h] / (denom[(size_t)d * HH + h] + 1e-16f);
  const float ea0 = ea[e*4+0], ea1 = ea[e*4+1], ea2 = ea[e*4+2], ea3 = ea[e*4+3];
  const float* vp = v + (size_t)s * HC + h * C;
  const float* wp = We + h * C;
  float* ap = agg + (size_t)d * HC + h * C;
  for (int c = lane; c < C; c += 32) {
    float ev = ea0*wp[c] + ea1*wp[HC+c] + ea2*wp[2*HC+c] + ea3*wp[3*HC+c];
    atomicAdd(&ap[c], (vp[c] + ev) * alpha);    // GLOBAL_ATOMIC_ADD_F32
  }
}

// ---------------------------------------------------------------------------
// Fused: head-mean -> beta gate (sigmoid of concat dot Wbeta) -> gated skip ->
// ELU -> LayerNorm.  One wave per node; values recomputed per pass to stay in
// registers (agg is L2-resident).
// ---------------------------------------------------------------------------
__global__ void __launch_bounds__(256) node_update(
    const float* __restrict__ agg, const float* __restrict__ rbuf,
    const float* __restrict__ Wbeta, const float* __restrict__ ln_g,
    const float* __restrict__ ln_b, float* __restrict__ hout,
    int Nn, int C) {
  const int lane = threadIdx.x & 31;
  const int node = blockIdx.x * 8 + (threadIdx.x >> 5);
  if (node >= Nn) return;
  const float* ag = agg  + (size_t)node * HH * C;
  const float* rp = rbuf + (size_t)node * C;

  float sb = 0.f;
  for (int c = lane; c < C; c += 32) {
    float o = 0.f;
    #pragma unroll
    for (int h = 0; h < HH; ++h) o += ag[h * C + c];
    o *= 0.125f;
    float r = rp[c];
    sb += o * Wbeta[c] + r * Wbeta[C + c] + (o - r) * Wbeta[2 * C + c];
  }
  for (int off = 16; off; off >>= 1) sb += __shfl_xor(sb, off, 32);
  const float beta = 1.f / (1.f + expf(-sb));

  float mu = 0.f;
  for (int c = lane; c < C; c += 32) {
    float o = 0.f;
    #pragma unroll
    for (int h = 0; h < HH; ++h) o += ag[h * C + c];
    o *= 0.125f;
    float xh = beta * rp[c] + (1.f - beta) * o;
    xh = xh > 0.f ? xh : expm1f(xh);
    mu += xh;
  }
  for (int off = 16; off; off >>= 1) mu += __shfl_xor(mu, off, 32);
  mu /= (float)C;

  float var = 0.f;
  for (int c = lane; c < C; c += 32) {
    float o = 0.f;
    #pragma unroll
    for (int h = 0; h < HH; ++h) o += ag[h * C + c];
    o *= 0.125f;
    float xh = beta * rp[c] + (1.f - beta) * o;
    xh = xh > 0.f ? xh : expm1f(xh);
    float dlt = xh - mu;
    var += dlt * dlt;
  }
  for (int off = 16; off; off >>= 1) var += __shfl_xor(var, off, 32);
  var /= (float)C;
  const float inv = rsqrtf(var + 1e-5f);

  for (int c = lane; c < C; c += 32) {
    float o = 0.f;
    #pragma unroll
    for (int h = 0; h < HH; ++h) o += ag[h * C + c];
    o *= 0.125f;
    float xh = beta * rp[c] + (1.f - beta) * o;
    xh = xh > 0.f ? xh : expm1f(xh);
    hout[(size_t)node * C + c] = (xh - mu) * inv * ln_g[c] + ln_b[c];
  }
}

// ---------------------------------------------------------------------------
// Global attention pooling + final FC (all tiny).
// ---------------------------------------------------------------------------
__global__ void __launch_bounds__(256) pool_gate(
    const float* __restrict__ h, const float* __restrict__ Wg,
    const float* __restrict__ bg, const int* __restrict__ batch,
    float* __restrict__ gate, unsigned* __restrict__ gmax, int Nn) {
  int n = blockIdx.x * 256 + threadIdx.x;
  if (n >= Nn) return;
  float s = bg[0];
  const float* hp = h + (size_t)n * 64;
  #pragma unroll 8
  for (int c = 0; c < 64; ++c) s += hp[c] * Wg[c];
  gate[n] = s;
  atomicMax(&gmax[batch[n]], f2ord(s));
}

__global__ void __launch_bounds__(256) pool_norm(
    float* __restrict__ gate, const unsigned* __restrict__ gmax,
    float* __restrict__ gden, const int* __restrict__ batch, int Nn) {
  int n = blockIdx.x * 256 + threadIdx.x;
  if (n >= Nn) return;
  int b = batch[n];
  float ex = expf(gate[n] - ord2f(gmax[b]));
  gate[n] = ex;
  atomicAdd(&gden[b], ex);
}

__global__ void __launch_bounds__(256) pool_scatter(
    const float* __restrict__ h, const float* __restrict__ gate,
    const float* __restrict__ gden, const int* __restrict__ batch,
    float* __restrict__ pooled, int Nn) {
  int i = blockIdx.x * 256 + threadIdx.x;
  if (i >= Nn * 64) return;
  int n = i >> 6, c = i & 63;
  int b = batch[n];
  float a = gate[n] / (gden[b] + 1e-16f);
  atomicAdd(&pooled[b * 64 + c], a * h[(size_t)n * 64 + c]);
}

__global__ void __launch_bounds__(256) final_fc(
    const float* __restrict__ pooled, const float* __restrict__ Wfc,
    const float* __restrict__ bfc, float* __restrict__ out) {
  int i = blockIdx.x * 256 + threadIdx.x;
  if (i >= GG * OUTD) return;
  int g = i / OUTD, o = i % OUTD;
  float s = bfc[o];
  #pragma unroll 8
  for (int c = 0; c < 64; ++c) s += pooled[g * 64 + c] * Wfc[c * OUTD + o];
  out[i] = s;
}

// ---------------------------------------------------------------------------
// Host orchestration.
// Input order: top-level setup_inputs dict order {x, edge_index, edge_attr,
// batch, params}; params flattened in JAX tree_flatten (alphabetical) order:
//   4:Wfc 5:Wg 6:bfc 7:bg, then per layer l (base=8+12l):
//   +0 We, +1 Wbeta, +2 Wk, +3 Wq, +4 Wskip, +5 Wv,
//   +6 bk, +7 bq, +8 bskip, +9 bv, +10 ln_b, +11 ln_g
// ---------------------------------------------------------------------------
extern "C" void kernel_launch(void* const* d_in, const int* in_sizes, int n_in,
                              void* d_out, int out_size, void* d_ws, size_t ws_size,
                              hipStream_t stream) {
  (void)in_sizes; (void)n_in; (void)out_size; (void)ws_size;
  const float* x    = (const float*)d_in[0];
  const int*   eidx = (const int*)d_in[1];
  const float* ea   = (const float*)d_in[2];
  const int*   batch= (const int*)d_in[3];
  const float* Wfc  = (const float*)d_in[4];
  const float* Wg   = (const float*)d_in[5];
  const float* bfc  = (const float*)d_in[6];
  const float* bg   = (const float*)d_in[7];
  const int* src  = eidx;
  const int* dstp = eidx + EE;

  float* ws = (float*)d_ws;
  size_t off = 0;
  float* qb   = ws + off; off += (size_t)NN * 4096;
  float* kb   = ws + off; off += (size_t)NN * 4096;
  float* vb   = ws + off; off += (size_t)NN * 4096;
  float* agg  = ws + off; off += (size_t)NN * 4096;
  float* logits = ws + off; off += (size_t)EE * HH;
  unsigned* nodemax = (unsigned*)(ws + off); off += (size_t)NN * HH;
  float* denom = ws + off; off += (size_t)NN * HH;
  float* rbuf  = ws + off; off += (size_t)NN * 512;
  float* hbuf  = ws + off; off += (size_t)NN * 512;
  float* gate  = ws + off; off += NN;
  unsigned* gmax = (unsigned*)(ws + off); off += GG;
  float* gden  = ws + off; off += GG;
  float* pooled = ws + off; off += GG * 64;

  const int DIN[3] = {32, 512, 256};
  const int CD[3]  = {512, 256, 64};

  for (int l = 0; l < 3; ++l) {
    const int din = DIN[l], C = CD[l], HC = HH * C;
    const int base = 8 + 12 * l;
    const float* We    = (const float*)d_in[base + 0];
    const float* Wbeta = (const float*)d_in[base + 1];
    const float* Wk    = (const float*)d_in[base + 2];
    const float* Wq    = (const float*)d_in[base + 3];
    const float* Wskip = (const float*)d_in[base + 4];
    const float* Wv    = (const float*)d_in[base + 5];
    const float* bk    = (const float*)d_in[base + 6];
    const float* bq    = (const float*)d_in[base + 7];
    const float* bskip = (const float*)d_in[base + 8];
    const float* bv    = (const float*)d_in[base + 9];
    const float* ln_b  = (const float*)d_in[base + 10];
    const float* ln_g  = (const float*)d_in[base + 11];
    const float* xin = (l == 0) ? x : hbuf;

    zero_f32<<<(NN * HC + 255) / 256, 256, 0, stream>>>(agg, NN * HC);
    zero_f32<<<(NN * HH + 255) / 256, 256, 0, stream>>>(denom, NN * HH);
    zero_f32<<<(NN * HH + 255) / 256, 256, 0, stream>>>((float*)nodemax, NN * HH);

    dim3 g1(HC / 64, NN / 128);
    gemm_wmma_f32<<<g1, 256, 0, stream>>>(xin, Wq, bq, qb, NN, HC, din);
    gemm_wmma_f32<<<g1, 256, 0, stream>>>(xin, Wk, bk, kb, NN, HC, din);
    gemm_wmma_f32<<<g1, 256, 0, stream>>>(xin, Wv, bv, vb, NN, HC, din);
    dim3 g2(C / 64, NN / 128);
    gemm_wmma_f32<<<g2, 256, 0, stream>>>(xin, Wskip, bskip, rbuf, NN, C, din);

    float scale = 1.0f / sqrtf((float)C);
    edge_logits<<<EE * HH / 8, 256, 0, stream>>>(qb, kb, ea, We, src, dstp,
                                                 logits, nodemax, C, HC, scale);
    edge_softmax<<<EE * HH / 256, 256, 0, stream>>>(logits, nodemax, denom,
                                                    dstp, EE * HH);
    edge_scatter<<<EE * HH / 8, 256, 0, stream>>>(vb, ea, We, src, dstp,
                                                  logits, denom, agg, C, HC);
    node_update<<<NN / 8, 256, 0, stream>>>(agg, rbuf, Wbeta, ln_g, ln_b,
                                            hbuf, NN, C);
  }

  zero_f32<<<1, 256, 0, stream>>>((float*)gmax, GG);
  zero_f32<<<1, 256, 0, stream>>>(gden, GG);
  zero_f32<<<(GG * 64 + 255) / 256, 256, 0, stream>>>(pooled, GG * 64);
  pool_gate<<<NN / 256, 256, 0, stream>>>(hbuf, Wg, bg, batch, gate, gmax, NN);
  pool_norm<<<NN / 256, 256, 0, stream>>>(gate, gmax, gden, batch, NN);
  pool_scatter<<<NN * 64 / 256, 256, 0, stream>>>(hbuf, gate, gden, batch, pooled, NN);
  final_fc<<<3, 256, 0, stream>>>(pooled, Wfc, bfc, (float*)d_out);
}